// Model_13950053777835
// MI455X (gfx1250) — compile-verified
//
#include <hip/hip_runtime.h>
#include <hip/hip_bf16.h>
#include <math.h>

// Problem constants (match reference)
#define VV   32000
#define HH   256
#define BB   16
#define TT   128
#define LL   50
#define GG   8
#define BETA 5.0f
#define GAMMA 2.0f
#define ATOL 1e-8f

typedef float v2f __attribute__((ext_vector_type(2)));
typedef float v8f __attribute__((ext_vector_type(8)));

static __device__ __forceinline__ float sigm(float x) { return 1.0f / (1.0f + __expf(-x)); }

// ---------------------------------------------------------------------------
// k_goal: goal_embed[b,h] = sum_j wv[g[b,j], h]
// ---------------------------------------------------------------------------
__global__ void k_goal(const int* __restrict__ g, const float* __restrict__ wv,
                       float* __restrict__ goal_embed) {
    int i = blockIdx.x * blockDim.x + threadIdx.x;   // 0..B*H
    if (i >= BB * HH) return;
    int b = i >> 8, h = i & 255;
    float s = 0.0f;
    for (int j = 0; j < GG; ++j)
        s += wv[(size_t)g[b * GG + j] * HH + h];
    goal_embed[i] = s;
}

// ---------------------------------------------------------------------------
// k_egather: E[b,l,h] = wv[ingr[b,l], h]
// ---------------------------------------------------------------------------
__global__ void k_egather(const int* __restrict__ ingr, const float* __restrict__ wv,
                          float* __restrict__ E) {
    int i = blockIdx.x * blockDim.x + threadIdx.x;   // 0..B*L*H
    if (i >= BB * LL * HH) return;
    int row = i / HH, h = i % HH;                    // row = b*L + l
    E[i] = wv[(size_t)ingr[row] * HH + h];
}

// ---------------------------------------------------------------------------
// k_flags: flags[t] = all(|wv[recipe[b,t],h]| <= ATOL)   (one block per t)
// ---------------------------------------------------------------------------
__global__ void k_flags(const int* __restrict__ recipe, const float* __restrict__ wv,
                        int* __restrict__ flags) {
    __shared__ int notall;
    int t = blockIdx.x;
    if (threadIdx.x == 0) notall = 0;
    __syncthreads();
    for (int i = threadIdx.x; i < BB * HH; i += blockDim.x) {
        int b = i >> 8, h = i & 255;
        float v = wv[(size_t)recipe[b * TT + t] * HH + h];
        if (fabsf(v) > ATOL) notall = 1;
    }
    __syncthreads();
    if (threadIdx.x == 0) flags[t] = (notall == 0) ? 1 : 0;
}

// ---------------------------------------------------------------------------
// k_init: ht0 = goal_embed @ U_g^T ; goal_term = goal_embed @ Y^T + y_bias
// ---------------------------------------------------------------------------
__global__ void k_init(const float* __restrict__ ge, const float* __restrict__ U_g,
                       const float* __restrict__ Y, const float* __restrict__ y_bias,
                       float* __restrict__ ht0, float* __restrict__ gterm) {
    int i = blockIdx.x * blockDim.x + threadIdx.x;
    if (i >= BB * HH) return;
    int b = i >> 8, h = i & 255;
    const float* gb = ge + b * HH;
    const float* ur = U_g + (size_t)h * HH;
    const float* yr = Y + (size_t)h * HH;
    float s0 = 0.0f, s1 = 0.0f;
    for (int k = 0; k < HH; ++k) { float gv = gb[k]; s0 += gv * ur[k]; s1 += gv * yr[k]; }
    ht0[i] = s0;
    gterm[i] = s1 + y_bias[h];
}

// ---------------------------------------------------------------------------
// WMMA helper: one 16x16 f32 tile, A row-major (16 x 256) in LDS, B from a
// per-lane weight row pointer (row-major, stride 256: B[k][n] = W[n*256+k]).
// lane layout per CDNA5 ISA: A frag v2f = A[M][k0+off .. +1], off = 2*lane[4];
// B frag v2f = W[n][k0+off .. +1]; C v8f: M = r + 8*lane[4], N = lane&15.
// ---------------------------------------------------------------------------
static __device__ __forceinline__ v8f wmma_row256(const float* __restrict__ ap,
                                                  const float* __restrict__ bp,
                                                  int off, v8f c) {
#pragma unroll 4
    for (int k0 = 0; k0 < HH; k0 += 4) {
        v2f av; av.x = ap[k0 + off]; av.y = ap[k0 + off + 1];
        v2f bv; bv.x = bp[k0 + off]; bv.y = bp[k0 + off + 1];
        c = __builtin_amdgcn_wmma_f32_16x16x4_f32(false, av, false, bv, (short)0, c,
                                                  false, false);
    }
    return c;
}

// ---------------------------------------------------------------------------
// k_rnn: persistent single-workgroup scan over T steps. 512 threads = 16 waves.
// Wave j owns hidden-column block j (16 columns of H=256).
// ---------------------------------------------------------------------------
__global__ __launch_bounds__(512, 1) void k_rnn(
    const int* __restrict__ recipe, const float* __restrict__ wv,
    const float* __restrict__ w_ih, const float* __restrict__ w_hh,
    const float* __restrict__ b_ih, const float* __restrict__ b_hh,
    const float* __restrict__ Z, const float* __restrict__ P,
    const float* __restrict__ Sm, const float* __restrict__ z_bias,
    const float* __restrict__ ht0, const float* __restrict__ gterm_g,
    const float* __restrict__ E, const int* __restrict__ flags,
    float* __restrict__ staged, float* __restrict__ out_ht,
    float* __restrict__ out_a, float* __restrict__ out_E) {
    __shared__ float xt[BB * HH], ht[BB * HH], htn[BB * HH], gt[BB * HH];
    __shared__ float tmp[BB * HH], tmpZ[BB * HH], hproj[BB * HH], ot[BB * HH];
    __shared__ float a_s[BB * LL], ap_s[BB * LL], d_s[BB * LL];
    __shared__ float an_s[BB * LL], au_s[BB * LL];
    __shared__ float ref_s[BB * 3];

    const int tid = threadIdx.x;
    const int lane = tid & 31;
    const int wave = tid >> 5;                 // 0..15
    const int nl = lane & 15;
    const int off = (lane >> 4) << 1;          // 0 or 2 (K-pair select)
    const int M = lane & 15;
    const int rsh = (lane >> 4) << 3;          // +8 rows for upper half-wave

    for (int i = tid; i < BB * HH; i += 512) { ht[i] = ht0[i]; gt[i] = gterm_g[i]; ot[i] = 0.0f; }
    for (int i = tid; i < BB * LL; i += 512) { a_s[i] = 0.0f; ap_s[i] = 0.0f; }
    __syncthreads();

    for (int t = 0; t < TT; ++t) {
        const int zf = flags[t];
        if (zf) {  // skipped step: outputs repeat previous ot; carries unchanged
            for (int i = tid; i < BB * HH; i += 512) {
                int b = i >> 8, h = i & 255;
                staged[((size_t)b * TT + t) * HH + h] = ot[i];
            }
            __syncthreads();
            continue;
        }
        // gather x_t
        for (int i = tid; i < BB * HH; i += 512) {
            int b = i >> 8, h = i & 255;
            xt[i] = wv[(size_t)recipe[b * TT + t] * HH + h];
        }
        // tmp[b,h] = sum_l (1 - a_prev[b,l]) * E[b,l,h]
        for (int i = tid; i < BB * HH; i += 512) {
            int b = i >> 8, h = i & 255;
            const float* Eb = E + (size_t)b * LL * HH + h;
            const float* apb = ap_s + b * LL;
            float s = 0.0f;
            for (int l = 0; l < LL; ++l) s += (1.0f - apb[l]) * Eb[l * HH];
            tmp[i] = s;
        }
        __syncthreads();

        // tmpZ = tmp @ Z^T + z_bias    (wave j -> tile j)
        {
            v8f c = {};
            c = wmma_row256(tmp + M * HH, Z + (size_t)(wave * 16 + nl) * HH, off, c);
            int h = wave * 16 + nl;
            float zb = z_bias[h];
#pragma unroll
            for (int r = 0; r < 8; ++r) tmpZ[(r + rsh) * HH + h] = c[r] + zb;
        }
        __syncthreads();

        // gates: gi = x_t @ w_ih^T, gh = ht @ w_hh^T, fused into ht2 -> htn
        {
            const float* ax = xt + M * HH;
            const float* ah = ht + M * HH;
            v8f ci[5], ch[5];
#pragma unroll
            for (int gate = 0; gate < 5; ++gate) {
                const float* bi = w_ih + (size_t)(gate * HH + wave * 16 + nl) * HH;
                const float* bh = w_hh + (size_t)(gate * HH + wave * 16 + nl) * HH;
                v8f a1 = {}, a2 = {};
#pragma unroll 4
                for (int k0 = 0; k0 < HH; k0 += 4) {
                    v2f xa; xa.x = ax[k0 + off]; xa.y = ax[k0 + off + 1];
                    v2f ha; ha.x = ah[k0 + off]; ha.y = ah[k0 + off + 1];
                    v2f b1; b1.x = bi[k0 + off]; b1.y = bi[k0 + off + 1];
                    v2f b2; b2.x = bh[k0 + off]; b2.y = bh[k0 + off + 1];
                    a1 = __builtin_amdgcn_wmma_f32_16x16x4_f32(false, xa, false, b1, (short)0, a1, false, false);
                    a2 = __builtin_amdgcn_wmma_f32_16x16x4_f32(false, ha, false, b2, (short)0, a2, false, false);
                }
                ci[gate] = a1; ch[gate] = a2;
            }
            int h = wave * 16 + nl;
            float bir = b_ih[0 * HH + h], bhr = b_hh[0 * HH + h];
            float biu = b_ih[1 * HH + h], bhu = b_hh[1 * HH + h];
            float bin = b_ih[2 * HH + h], bhn = b_hh[2 * HH + h];
            float big = b_ih[3 * HH + h], bhg = b_hh[3 * HH + h];
            float bii = b_ih[4 * HH + h], bhi = b_hh[4 * HH + h];
#pragma unroll
            for (int r = 0; r < 8; ++r) {
                int b = r + rsh;
                float i_r = ci[0][r] + bir, h_r = ch[0][r] + bhr;
                float i_u = ci[1][r] + biu, h_u = ch[1][r] + bhu;
                float i_n = ci[2][r] + bin, h_n = ch[2][r] + bhn;
                float i_g = ci[3][r] + big, h_g = ch[3][r] + bhg;
                float i_i = ci[4][r] + bii, h_i = ch[4][r] + bhi;
                float z_t = sigm(i_u + h_u);
                float r_t = sigm(i_r + h_r);
                float s_t = sigm(i_g + h_g);
                float q_t = sigm(i_i + h_i);
                float h_tilde = tanhf(i_n + r_t * h_n + s_t * gt[b * HH + h] + q_t * tmpZ[b * HH + h]);
                htn[b * HH + h] = h_tilde + z_t * (ht[b * HH + h] - h_tilde);
            }
        }
        __syncthreads();

        // h_proj = ht2 @ P^T  (wave j -> tile j)
        {
            v8f c = {};
            c = wmma_row256(htn + M * HH, P + (size_t)(wave * 16 + nl) * HH, off, c);
            int h = wave * 16 + nl;
#pragma unroll
            for (int r = 0; r < 8; ++r) hproj[(r + rsh) * HH + h] = c[r];
        }
        __syncthreads();

        // ref = softmax(BETA * ht2 @ S^T) over 3
        if (tid < 48) {
            int b = tid / 3, i = tid % 3;
            const float* hb = htn + b * HH;
            const float* sr = Sm + i * HH;
            float s = 0.0f;
            for (int h = 0; h < HH; ++h) s += hb[h] * sr[h];
            ref_s[b * 3 + i] = BETA * s;
        }
        __syncthreads();
        if (tid < 16) {
            float r0 = ref_s[tid * 3 + 0], r1 = ref_s[tid * 3 + 1], r2 = ref_s[tid * 3 + 2];
            float m = fmaxf(r0, fmaxf(r1, r2));
            float e0 = __expf(r0 - m), e1 = __expf(r1 - m), e2 = __expf(r2 - m);
            float inv = 1.0f / (e0 + e1 + e2);
            ref_s[tid * 3 + 0] = e0 * inv; ref_s[tid * 3 + 1] = e1 * inv; ref_s[tid * 3 + 2] = e2 * inv;
        }
        __syncthreads();

        // d[b,l] = h_proj[b,:] . E[b,l,:]
        for (int i = tid; i < BB * LL; i += 512) {
            int b = i / LL, l = i % LL;
            const float* hb = hproj + b * HH;
            const float* eb = E + ((size_t)b * LL + l) * HH;
            float s = 0.0f;
            for (int h = 0; h < HH; ++h) s += hb[h] * eb[h];
            d_s[i] = s;
        }
        __syncthreads();

        // alpha_n = softmax_l(GAMMA*(1-a)*d), alpha_u = softmax_l(GAMMA*a*d)
        if (tid < 16) {
            int b = tid;
            float mn = -1e30f, mu = -1e30f;
            for (int l = 0; l < LL; ++l) {
                float dv = d_s[b * LL + l], av = a_s[b * LL + l];
                float ln = GAMMA * (1.0f - av) * dv;
                float lu = GAMMA * av * dv;
                mn = fmaxf(mn, ln); mu = fmaxf(mu, lu);
            }
            float sn = 0.0f, su = 0.0f;
            for (int l = 0; l < LL; ++l) {
                float dv = d_s[b * LL + l], av = a_s[b * LL + l];
                float en = __expf(GAMMA * (1.0f - av) * dv - mn);
                float eu = __expf(GAMMA * av * dv - mu);
                an_s[b * LL + l] = en; au_s[b * LL + l] = eu;
                sn += en; su += eu;
            }
            float in_ = 1.0f / sn, iu_ = 1.0f / su;
            for (int l = 0; l < LL; ++l) { an_s[b * LL + l] *= in_; au_s[b * LL + l] *= iu_; }
        }
        __syncthreads();

        // out = ref0*h_proj + ref1*c_n + ref2*c_u ; write staged + ot
        for (int i = tid; i < BB * HH; i += 512) {
            int b = i >> 8, h = i & 255;
            const float* eb = E + (size_t)b * LL * HH + h;
            const float* an = an_s + b * LL;
            const float* au = au_s + b * LL;
            float cn = 0.0f, cu = 0.0f;
            for (int l = 0; l < LL; ++l) { float e = eb[l * HH]; cn += an[l] * e; cu += au[l] * e; }
            float o = ref_s[b * 3 + 0] * hproj[i] + ref_s[b * 3 + 1] * cn + ref_s[b * 3 + 2] * cu;
            ot[i] = o;
            staged[((size_t)b * TT + t) * HH + h] = o;
        }
        // carry updates: a_prev <- a (start-of-step), a <- a + ref1*alpha_n
        for (int i = tid; i < BB * LL; i += 512) {
            int b = i / LL;
            float av = a_s[i];
            ap_s[i] = av;
            a_s[i] = av + ref_s[b * 3 + 1] * an_s[i];
        }
        // ht <- ht2
        for (int i = tid; i < BB * HH; i += 512) ht[i] = htn[i];
        __syncthreads();
    }

    // tail outputs: ht, a, E_new = (1 - a_prev) * E
    for (int i = tid; i < BB * HH; i += 512) out_ht[i] = ht[i];
    for (int i = tid; i < BB * LL; i += 512) out_a[i] = a_s[i];
    for (int i = tid; i < BB * LL * HH; i += 512) {
        int bl = i / HH;
        out_E[i] = (1.0f - ap_s[bl]) * E[i];
    }
}

// ---------------------------------------------------------------------------
// k_logits: (2048 x 256) @ (256 x 32000) + fcb -> d_out, f32 WMMA.
// Register-blocked: each block owns a 64-row A macro-tile (LDS, 64 KB);
// each of 8 waves owns one 16-col N-tile and 4 M-subtiles, so every B
// fragment (the expensive 16-row gather from fcW) feeds 4 WMMAs.
// ---------------------------------------------------------------------------
#define MT 4            // M-subtiles per wave
__global__ __launch_bounds__(256, 2) void k_logits(const float* __restrict__ staged,
                                                   const float* __restrict__ fcW,
                                                   const float* __restrict__ fcb,
                                                   float* __restrict__ out) {
    __shared__ float As[16 * MT * HH];         // 64 rows x 256 k = 64 KB
    const int tid = threadIdx.x;
    const int lane = tid & 31;
    const int wave = tid >> 5;                 // 0..7
    const int m0 = blockIdx.y * (16 * MT);
    for (int i = tid; i < 16 * MT * HH; i += 256)
        As[i] = staged[(size_t)(m0 + (i >> 8)) * HH + (i & 255)];

    const int nl = lane & 15;
    const int off = (lane >> 4) << 1;
    const int rsh = (lane >> 4) << 3;
    const int n0 = (blockIdx.x * 8 + wave) * 16;
    const float* bp = fcW + (size_t)(n0 + nl) * HH;
    __builtin_prefetch(bp, 0, 1);              // global_prefetch_b8: pull lane's fcW row
    __syncthreads();

    const float* ap = As + (lane & 15) * HH;
    v8f c0 = {}, c1 = {}, c2 = {}, c3 = {};
#pragma unroll 4
    for (int k0 = 0; k0 < HH; k0 += 4) {
        v2f bv; bv.x = bp[k0 + off]; bv.y = bp[k0 + off + 1];
        v2f a0; a0.x = ap[k0 + off];            a0.y = ap[k0 + off + 1];
        v2f a1; a1.x = ap[16 * HH + k0 + off];  a1.y = ap[16 * HH + k0 + off + 1];
        v2f a2; a2.x = ap[32 * HH + k0 + off];  a2.y = ap[32 * HH + k0 + off + 1];
        v2f a3; a3.x = ap[48 * HH + k0 + off];  a3.y = ap[48 * HH + k0 + off + 1];
        c0 = __builtin_amdgcn_wmma_f32_16x16x4_f32(false, a0, false, bv, (short)0, c0, false, false);
        c1 = __builtin_amdgcn_wmma_f32_16x16x4_f32(false, a1, false, bv, (short)0, c1, false, false);
        c2 = __builtin_amdgcn_wmma_f32_16x16x4_f32(false, a2, false, bv, (short)0, c2, false, false);
        c3 = __builtin_amdgcn_wmma_f32_16x16x4_f32(false, a3, false, bv, (short)0, c3, false, false);
    }

    const int n = n0 + nl;
    const float bias = fcb[n];
    float* o0 = out + (size_t)(m0 + rsh) * VV + n;
#pragma unroll
    for (int r = 0; r < 8; ++r) o0[(size_t)(r +  0) * VV] = c0[r] + bias;
#pragma unroll
    for (int r = 0; r < 8; ++r) o0[(size_t)(r + 16) * VV] = c1[r] + bias;
#pragma unroll
    for (int r = 0; r < 8; ++r) o0[(size_t)(r + 32) * VV] = c2[r] + bias;
#pragma unroll
    for (int r = 0; r < 8; ++r) o0[(size_t)(r + 48) * VV] = c3[r] + bias;
}

// ---------------------------------------------------------------------------
// kernel_launch
// ---------------------------------------------------------------------------
extern "C" void kernel_launch(void* const* d_in, const int* in_sizes, int n_in,
                              void* d_out, int out_size, void* d_ws, size_t ws_size,
                              hipStream_t stream) {
    const int*   recipe = (const int*)d_in[0];
    const int*   g      = (const int*)d_in[1];
    const int*   ingr   = (const int*)d_in[2];
    const float* wv     = (const float*)d_in[3];
    const float* w_ih   = (const float*)d_in[4];
    const float* w_hh   = (const float*)d_in[5];
    const float* b_ih   = (const float*)d_in[6];
    const float* b_hh   = (const float*)d_in[7];
    const float* Z      = (const float*)d_in[8];
    const float* Y      = (const float*)d_in[9];
    const float* U_g    = (const float*)d_in[10];
    const float* z_bias = (const float*)d_in[11];
    const float* y_bias = (const float*)d_in[12];
    const float* S      = (const float*)d_in[13];
    const float* P      = (const float*)d_in[14];
    const float* fcW    = (const float*)d_in[15];
    const float* fcb    = (const float*)d_in[16];

    float* ws = (float*)d_ws;
    float* goal_embed = ws;                       // 4096
    float* ht0        = ws + 4096;                // 4096
    float* gterm      = ws + 8192;                // 4096
    float* E          = ws + 12288;               // 204800
    float* staged     = ws + 12288 + 204800;      // 524288
    int*   flags      = (int*)(staged + 524288);  // 128

    float* out   = (float*)d_out;
    float* o_ht  = out + (size_t)BB * TT * VV;    // 4096
    float* o_a   = o_ht + BB * HH;                // 800
    float* o_E   = o_a + BB * LL;                 // 204800

    k_goal<<<(BB * HH + 255) / 256, 256, 0, stream>>>(g, wv, goal_embed);
    k_egather<<<(BB * LL * HH + 255) / 256, 256, 0, stream>>>(ingr, wv, E);
    k_flags<<<TT, 256, 0, stream>>>(recipe, wv, flags);
    k_init<<<(BB * HH + 255) / 256, 256, 0, stream>>>(goal_embed, U_g, Y, y_bias, ht0, gterm);

    k_rnn<<<1, 512, 0, stream>>>(recipe, wv, w_ih, w_hh, b_ih, b_hh, Z, P, S, z_bias,
                                 ht0, gterm, E, flags, staged, o_ht, o_a, o_E);

    k_logits<<<dim3(VV / (16 * 8), (BB * TT) / (16 * MT)), 256, 0, stream>>>(staged, fcW, fcb, out);
}